// CachedMultiHeadAttention_2035814498429
// MI455X (gfx1250) — compile-verified
//
#include <hip/hip_runtime.h>
#include <hip/hip_bf16.h>
#include <math.h>

#define EMBED 4096
#define NHEAD 32
#define HSIZE 128
#define KVLEN 8191
#define LNEW  8192   // KVLEN + 1

typedef float v2f __attribute__((ext_vector_type(2)));
typedef float v8f __attribute__((ext_vector_type(8)));

// ---------------------------------------------------------------------------
// GEMV  y = W x + b  (W: 4096x4096 row-major, y[o] = dot(W[o,:], x))
// via V_WMMA_F32_16X16X4_F32.
//   Per wave: 16 output rows. A = 16x4 tile of W:
//     lanes 0-15  hold M=lane,    K = {k, k+1}   (v2f load at W[row] + k)
//     lanes 16-31 hold M=lane-16, K = {k+2, k+3} (v2f load at W[row] + k + 2)
//   B = x chunk replicated across all 16 columns:
//     v0: lanes0-15 = x[k],  lanes16-31 = x[k+2];  v1: x[k+1] / x[k+3]
//     => same "base + 2*half" indexing as A, read from LDS.
//   Every column of D then equals W_tile . x; lanes with N==0 (lane 0 and 16)
//   hold the 16 results across their 8 accumulator VGPRs.
// ---------------------------------------------------------------------------
__global__ __launch_bounds__(128) void gemv_wmma_kernel(
    const float* __restrict__ x,
    const float* __restrict__ W0, const float* __restrict__ b0, float* __restrict__ o0,
    const float* __restrict__ W1, const float* __restrict__ b1, float* __restrict__ o1,
    const float* __restrict__ W2, const float* __restrict__ b2, float* __restrict__ o2)
{
    __shared__ float xs[EMBED];
    const int tid = threadIdx.x;

    for (int i = tid; i < EMBED / 4; i += 128)
        ((float4*)xs)[i] = ((const float4*)x)[i];
    __syncthreads();

    const float* W = W0; const float* b = b0; float* o = o0;
    if (blockIdx.y == 1)      { W = W1; b = b1; o = o1; }
    else if (blockIdx.y == 2) { W = W2; b = b2; o = o2; }

    const int wave = tid >> 5;
    const int lane = tid & 31;
    const int half = lane >> 4;   // 0 -> K mod 4 in {0,1}; 1 -> {2,3}
    const int m    = lane & 15;   // row within 16-row tile
    const int rowBase = blockIdx.x * 64 + wave * 16;

    const float* wrow = W + (size_t)(rowBase + m) * EMBED + 2 * half;
    const float* xk   = xs + 2 * half;

    v8f acc = {0.f, 0.f, 0.f, 0.f, 0.f, 0.f, 0.f, 0.f};

    for (int k = 0; k < EMBED; k += 16) {
        v2f a0 = *(const v2f*)(wrow + k);
        v2f a1 = *(const v2f*)(wrow + k + 4);
        v2f a2 = *(const v2f*)(wrow + k + 8);
        v2f a3 = *(const v2f*)(wrow + k + 12);
        v2f q0 = *(const v2f*)(xk + k);
        v2f q1 = *(const v2f*)(xk + k + 4);
        v2f q2 = *(const v2f*)(xk + k + 8);
        v2f q3 = *(const v2f*)(xk + k + 12);
        acc = __builtin_amdgcn_wmma_f32_16x16x4_f32(false, a0, false, q0, (short)0, acc, false, false);
        acc = __builtin_amdgcn_wmma_f32_16x16x4_f32(false, a1, false, q1, (short)0, acc, false, false);
        acc = __builtin_amdgcn_wmma_f32_16x16x4_f32(false, a2, false, q2, (short)0, acc, false, false);
        acc = __builtin_amdgcn_wmma_f32_16x16x4_f32(false, a3, false, q3, (short)0, acc, false, false);
    }

    // Column N==0 lives in lanes 0 (rows 0..7) and 16 (rows 8..15).
    if (m == 0) {
        const int base = rowBase + (half ? 8 : 0);
        #pragma unroll
        for (int r = 0; r < 8; ++r)
            o[base + r] = acc[r] + b[base + r];
    }
}

// ---------------------------------------------------------------------------
// Attention: one block per head. Scores for all 8192 rows kept in LDS (32 KB),
// single pass over K, LDS tree reductions, then coalesced AV pass over V.
// ---------------------------------------------------------------------------
__global__ __launch_bounds__(256) void attn_kernel(
    const float* __restrict__ q, const float* __restrict__ kc,
    const float* __restrict__ vc, float* __restrict__ o)
{
    __shared__ float sc[LNEW];
    __shared__ float qh[HSIZE];
    __shared__ float red[256];

    const int h   = blockIdx.x;
    const int tid = threadIdx.x;
    const size_t hbase = (size_t)h * HSIZE;

    // q head, pre-scaled by 1/sqrt(128)
    if (tid < HSIZE) qh[tid] = q[hbase + tid] * 0.0883883476483184f;
    __syncthreads();

    // ---- scores: each thread owns rows l = tid + 256*j --------------------
    float lmax = -3.402823466e+38f;
    for (int j = 0; j < LNEW / 256; ++j) {
        const int l = tid + 256 * j;
        const float* kr = kc + ((size_t)l << 12) + hbase;
        float s = 0.f;
        #pragma unroll
        for (int i = 0; i < HSIZE; i += 4) {
            const float4 kv = *(const float4*)(kr + i);
            s += kv.x * qh[i] + kv.y * qh[i + 1] + kv.z * qh[i + 2] + kv.w * qh[i + 3];
        }
        sc[l] = s;
        lmax = fmaxf(lmax, s);
    }

    // ---- block max --------------------------------------------------------
    red[tid] = lmax;
    __syncthreads();
    for (int off = 128; off > 0; off >>= 1) {
        if (tid < off) red[tid] = fmaxf(red[tid], red[tid + off]);
        __syncthreads();
    }
    const float mx = red[0];
    __syncthreads();

    // ---- exp + block sum --------------------------------------------------
    float lsum = 0.f;
    for (int j = 0; j < LNEW / 256; ++j) {
        const int l = tid + 256 * j;
        const float p = __expf(sc[l] - mx);
        sc[l] = p;
        lsum += p;
    }
    red[tid] = lsum;
    __syncthreads();
    for (int off = 128; off > 0; off >>= 1) {
        if (tid < off) red[tid] += red[tid + off];
        __syncthreads();
    }
    const float inv = 1.0f / red[0];
    __syncthreads();

    // ---- AV: thread (d, parity) accumulates over half the rows ------------
    const int d  = tid & (HSIZE - 1);
    const int r0 = tid >> 7;            // 0 or 1
    const float* vcol = vc + hbase + d;
    float a0 = 0.f, a1 = 0.f, a2 = 0.f, a3 = 0.f;
    for (int l = r0; l < LNEW; l += 8) {
        a0 += sc[l]     * vcol[(size_t)l << 12];
        a1 += sc[l + 2] * vcol[(size_t)(l + 2) << 12];
        a2 += sc[l + 4] * vcol[(size_t)(l + 4) << 12];
        a3 += sc[l + 6] * vcol[(size_t)(l + 6) << 12];
    }
    red[tid] = (a0 + a1) + (a2 + a3);
    __syncthreads();
    if (tid < HSIZE) o[hbase + d] = (red[tid] + red[tid + HSIZE]) * inv;
}

// ---------------------------------------------------------------------------
extern "C" void kernel_launch(void* const* d_in, const int* in_sizes, int n_in,
                              void* d_out, int out_size, void* d_ws, size_t ws_size,
                              hipStream_t stream) {
    (void)in_sizes; (void)n_in; (void)out_size; (void)ws_size;

    const float* seq      = (const float*)d_in[0];
    const float* k_cached = (const float*)d_in[1];
    const float* v_cached = (const float*)d_in[2];
    const float* Wq_w     = (const float*)d_in[3];
    const float* Wq_b     = (const float*)d_in[4];
    const float* Wk_w     = (const float*)d_in[5];
    const float* Wk_b     = (const float*)d_in[6];
    const float* Wv_w     = (const float*)d_in[7];
    const float* Wv_b     = (const float*)d_in[8];

    float* out   = (float*)d_out;                     // [4096]
    float* k_new = out + EMBED;                       // [8192 x 4096]
    float* v_new = k_new + (size_t)LNEW * EMBED;      // [8192 x 4096]

    float* q_ws    = (float*)d_ws;                    // [4096]
    float* attn_ws = q_ws + EMBED;                    // [4096]

    // 1) KV cache append (bulk copy): graph-capture-safe d2d async copies.
    hipMemcpyAsync(k_new, k_cached, (size_t)KVLEN * EMBED * sizeof(float),
                   hipMemcpyDeviceToDevice, stream);
    hipMemcpyAsync(v_new, v_cached, (size_t)KVLEN * EMBED * sizeof(float),
                   hipMemcpyDeviceToDevice, stream);

    // 2) Q/K/V projections in one launch (blockIdx.y selects the matrix).
    gemv_wmma_kernel<<<dim3(64, 3), 128, 0, stream>>>(
        seq,
        Wq_w, Wq_b, q_ws,
        Wk_w, Wk_b, k_new + (size_t)KVLEN * EMBED,
        Wv_w, Wv_b, v_new + (size_t)KVLEN * EMBED);

    // 3) Attention: one block per head.
    attn_kernel<<<NHEAD, 256, 0, stream>>>(q_ws, k_new, v_new, attn_ws);

    // 4) Output projection (reference reuses Wq/Wq_b).
    gemv_wmma_kernel<<<dim3(64, 1), 128, 0, stream>>>(
        attn_ws,
        Wq_w, Wq_b, out,
        nullptr, nullptr, nullptr,
        nullptr, nullptr, nullptr);
}